// LTCCell_46119358825197
// MI455X (gfx1250) — compile-verified
//
#include <hip/hip_runtime.h>

#define DEVI __device__ __forceinline__

#define B_N 256
#define I_N 512
#define H_N 512
#define K_N (I_N + H_N)       // 1024 total reduction length (sensory then inter)
#define HT 32                 // h-tile per block (== lanes per wave)
#define BT 16                 // b-tile per block
#define NB 2                  // batches per thread
#define KC 32                 // k chunk staged in LDS
#define NCHUNK (K_N / KC)     // 32
#define SENS_CHUNKS (I_N / KC)// 16

// ---------------- fast tanh (probe gfx1250 v_tanh_f32) ----------------
#if __has_builtin(__builtin_amdgcn_tanhf)
DEVI float ftanh(float x) { return __builtin_amdgcn_tanhf(x); }
#elif __has_builtin(__builtin_amdgcn_tanh_f32)
DEVI float ftanh(float x) { return __builtin_amdgcn_tanh_f32(x); }
#else
// tanh(x) = 2*sigmoid(2x)-1 = 2/(1+exp2(-2x*log2e)) - 1
DEVI float ftanh(float x) {
  float e = __builtin_amdgcn_exp2f(x * -2.8853900817779268f);
  return __builtin_amdgcn_rcpf(1.0f + e) * 2.0f - 1.0f;
}
#endif

// ---------------- async global->LDS copies (gfx1250) ----------------
#if __has_builtin(__builtin_amdgcn_global_load_async_to_lds_b128) && \
    __has_builtin(__builtin_amdgcn_global_load_async_to_lds_b32)
#define LTC_ASYNC 1
#endif

typedef int ltc_v4i __attribute__((ext_vector_type(4)));
typedef __attribute__((address_space(1))) ltc_v4i* ltc_g128p;
typedef __attribute__((address_space(3))) ltc_v4i* ltc_l128p;
typedef __attribute__((address_space(1))) int* ltc_g32p;
typedef __attribute__((address_space(3))) int* ltc_l32p;

DEVI void cp16(void* l, const void* g) {
#ifdef LTC_ASYNC
  __builtin_amdgcn_global_load_async_to_lds_b128((ltc_g128p)g, (ltc_l128p)l, 0, 0);
#else
  *(float4*)l = *(const float4*)g;
#endif
}
DEVI void cp4(void* l, const void* g) {
#ifdef LTC_ASYNC
  __builtin_amdgcn_global_load_async_to_lds_b32((ltc_g32p)g, (ltc_l32p)l, 0, 0);
#else
  *(float*)l = *(const float*)g;
#endif
}

template <int N>
DEVI void wait_async() {
#ifdef LTC_ASYNC
#if __has_builtin(__builtin_amdgcn_s_wait_asynccnt)
  __builtin_amdgcn_s_wait_asynccnt(N);
#else
  asm volatile("s_wait_asynccnt %0" ::"i"(N));
#endif
#endif
}

// ---------------- main LTC kernel (defined FIRST for disasm visibility) ----------------
// Inner math per synapse (k,h,b): u = fma(x, A, B); t = tanh(u);
//   den += Wd*t; num += Wn*t      (constant +Wd/+Wn terms hoisted into column sums)
__global__ __launch_bounds__(256) void ltc_main_kernel(
    const float* __restrict__ inputs, const float* __restrict__ state,
    const float4* __restrict__ P,
    const float* __restrict__ csum_d, const float* __restrict__ csum_n,
    const float* __restrict__ vleak, const float* __restrict__ gleak,
    const float* __restrict__ cm, float* __restrict__ out) {
  __shared__ float4 wbuf[2][KC * HT];  // 2 * 32*32 * 16B = 32 KB
  __shared__ float xbuf[2][KC * BT];   // 2 * 32*16 *  4B =  4 KB, layout x[kk][b_local]

  const int tid = threadIdx.x;
  const int lane = tid & 31;  // -> h within tile
  const int wv = tid >> 5;    // wave id 0..7 -> batch slot
  const int h = blockIdx.x * HT + lane;
  const int b0 = blockIdx.y * BT + wv * NB;

  auto issue = [&](int c, int s) {
    const int k0 = c * KC;
    // weight chunk: 32 k-rows x 32 h-cols of float4 -> 4 async b128 per thread
    const float4* wsrc = P + (size_t)k0 * H_N + (size_t)blockIdx.x * HT;
    const int col = tid & 31;
    const int r0 = tid >> 5;
#pragma unroll
    for (int q = 0; q < 4; ++q) {
      int r = r0 + 8 * q;
      cp16(&wbuf[s][r * HT + col], wsrc + (size_t)r * H_N + col);
    }
    // x chunk, transposed into LDS as x[kk][b_local]: 2 async b32 per thread
    const float* xsrc = (c < SENS_CHUNKS) ? inputs : state;
    const int kx = (c < SENS_CHUNKS) ? k0 : (k0 - I_N);
#pragma unroll
    for (int q = 0; q < 2; ++q) {
      int e = tid + 256 * q;   // [0,512)
      int kk = e >> 4;         // [0,32)
      int bl = e & 15;         // [0,16)
      int bg = blockIdx.y * BT + bl;
      cp4(&xbuf[s][kk * BT + bl], xsrc + (size_t)bg * 512 + kx + kk);
    }
  };

  issue(0, 0);

  float den0 = 0.f, num0 = 0.f, den1 = 0.f, num1 = 0.f;

  for (int c = 0; c < NCHUNK; ++c) {
    const int cur = c & 1;
    if (c + 1 < NCHUNK) {
      issue(c + 1, cur ^ 1);
      wait_async<6>();  // 6 newest in flight -> previous chunk's 6 complete (in-order)
    } else {
      wait_async<0>();
    }
    __syncthreads();
#pragma unroll
    for (int kk = 0; kk < KC; ++kk) {
      float4 w4 = wbuf[cur][kk * HT + lane];              // lane-strided b128
      float x0 = xbuf[cur][kk * BT + wv * NB + 0];        // wave-uniform broadcast
      float x1 = xbuf[cur][kk * BT + wv * NB + 1];
      float t0 = ftanh(fmaf(x0, w4.x, w4.y));
      float t1 = ftanh(fmaf(x1, w4.x, w4.y));
      den0 = fmaf(w4.z, t0, den0);
      num0 = fmaf(w4.w, t0, num0);
      den1 = fmaf(w4.z, t1, den1);
      num1 = fmaf(w4.w, t1, num1);
    }
    __syncthreads();
  }

  // epilogue
  const float eps = 1e-8f;
  const float csd = csum_d[h], csn = csum_n[h];
  const float gl = gleak[h], vl = vleak[h], cmh = cm[h];
#pragma unroll
  for (int j = 0; j < NB; ++j) {
    const int b = b0 + j;
    const float wnum = (j == 0 ? num0 : num1) + csn;
    const float wden = (j == 0 ? den0 : den1) + csd;
    const float st = state[(size_t)b * H_N + h];
    const float G = gl + wden;
    const float tau = cmh / (G + eps);
    const float numer = fmaf(cmh, st, fmaf(gl, vl, wnum));
    const float denom = cmh + G + eps;
    const float vinf = numer / denom;
    // exp(-dt/(tau+eps)) via exp2
    const float efac = __builtin_amdgcn_exp2f((-0.1f * 1.4426950408889634f) / (tau + eps));
    const float ns = vinf + (st - vinf) * efac;
    out[(size_t)b * H_N + h] = ftanh(ns);
  }
}

// ---------------- pack params ----------------
// P[k*H+h] = (A, B, Wd, Wn) with A=0.5*sigma, B=-0.5*sigma*mu, Wd=0.5*W, Wn=0.5*W*erev
// k in [0,I) -> sensory params, k in [I,I+H) -> inter params.
__global__ void ltc_prep_kernel(const float* __restrict__ smu, const float* __restrict__ ssig,
                                const float* __restrict__ sW, const float* __restrict__ serev,
                                const float* __restrict__ imu, const float* __restrict__ isig,
                                const float* __restrict__ iW, const float* __restrict__ ierev,
                                float4* __restrict__ P) {
  int idx = blockIdx.x * blockDim.x + threadIdx.x;
  if (idx >= K_N * H_N) return;
  const int n1 = I_N * H_N;
  float mu, sg, w, er;
  if (idx < n1) {
    mu = smu[idx]; sg = ssig[idx]; w = sW[idx]; er = serev[idx];
  } else {
    int j = idx - n1;
    mu = imu[j]; sg = isig[j]; w = iW[j]; er = ierev[j];
  }
  float4 p;
  p.x = 0.5f * sg;
  p.y = -0.5f * sg * mu;
  p.z = 0.5f * w;
  p.w = 0.5f * w * er;
  P[idx] = p;
}

// ---------------- batch-independent column sums ----------------
__global__ void ltc_colsum_kernel(const float4* __restrict__ P,
                                  float* __restrict__ csum_d, float* __restrict__ csum_n) {
  int h = blockIdx.x * blockDim.x + threadIdx.x;
  if (h >= H_N) return;
  float sd = 0.f, sn = 0.f;
  for (int k = 0; k < K_N; ++k) {
    float4 p = P[(size_t)k * H_N + h];
    sd += p.z;
    sn += p.w;
  }
  csum_d[h] = sd;
  csum_n[h] = sn;
}

extern "C" void kernel_launch(void* const* d_in, const int* in_sizes, int n_in,
                              void* d_out, int out_size, void* d_ws, size_t ws_size,
                              hipStream_t stream) {
  (void)in_sizes; (void)n_in; (void)out_size; (void)ws_size;

  const float* inputs = (const float*)d_in[0];
  const float* state  = (const float*)d_in[1];
  const float* smu    = (const float*)d_in[2];
  const float* ssig   = (const float*)d_in[3];
  const float* sW     = (const float*)d_in[4];
  const float* serev  = (const float*)d_in[5];
  const float* imu    = (const float*)d_in[6];
  const float* isig   = (const float*)d_in[7];
  const float* iW     = (const float*)d_in[8];
  const float* ierev  = (const float*)d_in[9];
  const float* vleak  = (const float*)d_in[10];
  const float* gleak  = (const float*)d_in[11];
  const float* cm     = (const float*)d_in[12];

  float4* P      = (float4*)d_ws;                                      // 8 MB
  float* csum_d  = (float*)((char*)d_ws + (size_t)K_N * H_N * sizeof(float4));
  float* csum_n  = csum_d + H_N;

  ltc_prep_kernel<<<(K_N * H_N) / 256, 256, 0, stream>>>(
      smu, ssig, sW, serev, imu, isig, iW, ierev, P);
  ltc_colsum_kernel<<<H_N / 256, 256, 0, stream>>>(P, csum_d, csum_n);
  ltc_main_kernel<<<dim3(H_N / HT, B_N / BT), 256, 0, stream>>>(
      inputs, state, P, csum_d, csum_n, vleak, gleak, cm, (float*)d_out);
}